// MGEmbedder_24103356465172
// MI455X (gfx1250) — compile-verified
//
#include <hip/hip_runtime.h>

typedef float v2f __attribute__((ext_vector_type(2)));
typedef float v4f __attribute__((ext_vector_type(4)));
typedef float v8f __attribute__((ext_vector_type(8)));

#define TT    4
#define NG    12288
#define CC    16
#define NHB   9
#define FF    144      // K = NH*C
#define NO    1024     // N = UP*E
#define FP    148      // padded LDS row stride (floats) -> bank-conflict-free
#define NS    36       // K / 4 WMMA k-steps
#define WP_BYTES ((size_t)NO * FF * 4)   // 589824 B packed-W scratch

// ---------------------------------------------------------------------------
// Prep: pack W (144x1024 row-major) into WMMA-B fragment order:
//   Wp[(n*2 + h)*72 + 2*s + w] = W[4*s + 2*h + w][n]
// so a lane (column n, half h) reads k-step s as one contiguous float2, and
// two consecutive k-steps as one 16B-aligned float4.
// ---------------------------------------------------------------------------
__global__ void pack_W_kernel(const float* __restrict__ W, float* __restrict__ Wp) {
    int idx = blockIdx.x * 256 + threadIdx.x;        // over 144*1024
    if (idx < FF * NO) {
        int k = idx / NO;
        int n = idx % NO;
        int s = k >> 2;
        int h = (k >> 1) & 1;
        int w = k & 1;
        Wp[(size_t)(n * 2 + h) * 72 + 2 * s + w] = W[idx];
    }
}

// ---------------------------------------------------------------------------
// Fast path: block = 8 waves, 64 rows x 512 cols. Wave = 4 M-tiles x 64 cols,
// 16 f32 WMMA accumulators. B from packed Wp via global_load_b128.
// ---------------------------------------------------------------------------
__global__ __launch_bounds__(256, 1)
void mg_gemm_packed(const float* __restrict__ mg_emb,
                    const int*   __restrict__ var_indices,
                    const int*   __restrict__ adjc,
                    const float* __restrict__ Wp,
                    const float* __restrict__ bias,
                    float*       __restrict__ out)
{
    __shared__ float xs[64 * FP];

    const int tid      = threadIdx.x;
    const int row_blk  = blockIdx.x >> 1;
    const int col_half = blockIdx.x & 1;
    const int m0       = row_blk * 64;               // 64 | NG -> fixed (v,t)
    const int v        = m0 / (TT * NG);
    const int t        = (m0 / NG) % TT;
    const int p0       = m0 % NG;
    const int vi       = var_indices[v];

    const float* embT = mg_emb + (size_t)(vi * TT + t) * NG * CC;

    // ---- Gather A tile: 64 rows x 9 neighbors x 4 float4s ----
    for (int idx = tid; idx < 64 * NHB * 4; idx += 256) {
        const int r   = idx / (NHB * 4);
        const int rem = idx % (NHB * 4);
        const int h   = rem >> 2;
        const int cg  = rem & 3;
        const int nbr = adjc[(p0 + r) * NHB + h];
        const float4 val = *(const float4*)(embT + (size_t)nbr * CC + cg * 4);
        float* dst = &xs[r * FP + h * CC + cg * 4];
        dst[0] = val.x; dst[1] = val.y; dst[2] = val.z; dst[3] = val.w;
    }
    __syncthreads();

    const int wave   = tid >> 5;
    const int lane   = tid & 31;
    const int lcol   = lane & 15;
    const int khalf  = lane >> 4;
    const int n_base = col_half * 512 + wave * 64;

    // Per-nb packed-W base pointers (column = n_base + nb*16 + lcol)
    const float* wb[4];
#pragma unroll
    for (int nb = 0; nb < 4; ++nb) {
        const int col = n_base + nb * 16 + lcol;
        wb[nb] = Wp + (size_t)(col * 2 + khalf) * 72;
    }

    // Accumulators init with broadcast bias
    v8f acc[4][4];
#pragma unroll
    for (int nb = 0; nb < 4; ++nb) {
        const float bv = bias[n_base + nb * 16 + lcol];
#pragma unroll
        for (int mt = 0; mt < 4; ++mt)
            acc[mt][nb] = (v8f){bv, bv, bv, bv, bv, bv, bv, bv};
    }

    const float* xrow[4];
#pragma unroll
    for (int mt = 0; mt < 4; ++mt)
        xrow[mt] = &xs[(mt * 16 + lcol) * FP];

    // ---- K loop: 18 iterations of 2 k-steps (K=8), 32 WMMAs each ----
    for (int s = 0; s < NS; s += 2) {
        const int kb = 4 * s + 2 * khalf;
        v2f a0[4], a1[4];
#pragma unroll
        for (int mt = 0; mt < 4; ++mt) {
            a0[mt] = *(const v2f*)(xrow[mt] + kb);       // k-step s
            a1[mt] = *(const v2f*)(xrow[mt] + kb + 4);   // k-step s+1
        }
#pragma unroll
        for (int nb = 0; nb < 4; ++nb) {
            const v4f w4 = *(const v4f*)(wb[nb] + 2 * s);  // 2 B frags, one b128
            const v2f b0 = {w4.x, w4.y};
            const v2f b1 = {w4.z, w4.w};
#pragma unroll
            for (int mt = 0; mt < 4; ++mt)
                acc[mt][nb] = __builtin_amdgcn_wmma_f32_16x16x4_f32(
                    false, a0[mt], false, b0, (short)0, acc[mt][nb], false, false);
#pragma unroll
            for (int mt = 0; mt < 4; ++mt)
                acc[mt][nb] = __builtin_amdgcn_wmma_f32_16x16x4_f32(
                    false, a1[mt], false, b1, (short)0, acc[mt][nb], false, false);
        }
    }

    // ---- Store ----
    const int rbase = khalf * 8;
#pragma unroll
    for (int mt = 0; mt < 4; ++mt) {
#pragma unroll
        for (int nb = 0; nb < 4; ++nb) {
            const int col = n_base + nb * 16 + lcol;
#pragma unroll
            for (int i = 0; i < 8; ++i)
                out[(size_t)(m0 + mt * 16 + rbase + i) * NO + col] = acc[mt][nb][i];
        }
    }
}

// ---------------------------------------------------------------------------
// Fallback (no scratch): round-1 kernel, direct W reads.
// ---------------------------------------------------------------------------
__global__ __launch_bounds__(256, 1)
void mg_gemm_direct(const float* __restrict__ mg_emb,
                    const int*   __restrict__ var_indices,
                    const int*   __restrict__ adjc,
                    const float* __restrict__ W,
                    const float* __restrict__ bias,
                    float*       __restrict__ out)
{
    __shared__ float xs[16 * FP];

    const int tid = threadIdx.x;
    const int m0  = blockIdx.x * 16;
    const int v   = m0 / (TT * NG);
    const int t   = (m0 / NG) % TT;
    const int p0  = m0 % NG;
    const int vi  = var_indices[v];
    const float* embT = mg_emb + (size_t)(vi * TT + t) * NG * CC;

    for (int idx = tid; idx < 16 * NHB * 4; idx += 256) {
        const int r   = idx / (NHB * 4);
        const int rem = idx % (NHB * 4);
        const int h   = rem >> 2;
        const int cg  = rem & 3;
        const int nbr = adjc[(p0 + r) * NHB + h];
        const float4 val = *(const float4*)(embT + (size_t)nbr * CC + cg * 4);
        float* dst = &xs[r * FP + h * CC + cg * 4];
        dst[0] = val.x; dst[1] = val.y; dst[2] = val.z; dst[3] = val.w;
    }
    __syncthreads();

    const int wave   = tid >> 5;
    const int lane   = tid & 31;
    const int lcol   = lane & 15;
    const int khalf  = lane >> 4;
    const int n_base = wave * 128;

    v8f acc[8];
#pragma unroll
    for (int nb = 0; nb < 8; ++nb) {
        const float bv = bias[n_base + nb * 16 + lcol];
        acc[nb] = (v8f){bv, bv, bv, bv, bv, bv, bv, bv};
    }

    const float* xrow = &xs[lcol * FP];
    for (int k0 = 0; k0 < FF; k0 += 4) {
        const int k = k0 + 2 * khalf;
        const v2f a = *(const v2f*)(xrow + k);
        const float* wk = W + (size_t)k * NO;
#pragma unroll
        for (int nb = 0; nb < 8; ++nb) {
            const int col = n_base + nb * 16 + lcol;
            v2f bf;
            bf.x = wk[col];
            bf.y = wk[NO + col];
            acc[nb] = __builtin_amdgcn_wmma_f32_16x16x4_f32(
                false, a, false, bf, (short)0, acc[nb], false, false);
        }
    }

    const int rbase = khalf * 8;
#pragma unroll
    for (int nb = 0; nb < 8; ++nb) {
        const int col = n_base + nb * 16 + lcol;
#pragma unroll
        for (int i = 0; i < 8; ++i)
            out[(size_t)(m0 + rbase + i) * NO + col] = acc[nb][i];
    }
}

extern "C" void kernel_launch(void* const* d_in, const int* in_sizes, int n_in,
                              void* d_out, int out_size, void* d_ws, size_t ws_size,
                              hipStream_t stream) {
    (void)in_sizes; (void)n_in; (void)out_size;
    const float* mg_emb      = (const float*)d_in[0];
    const int*   var_indices = (const int*)  d_in[1];
    const int*   adjc        = (const int*)  d_in[2];
    const float* W           = (const float*)d_in[3];
    const float* bias        = (const float*)d_in[4];
    float*       out         = (float*)d_out;

    const int M = 1 * 4 * TT * NG;   // 196608 rows

    if (d_ws != nullptr && ws_size >= WP_BYTES) {
        float* Wp = (float*)d_ws;
        hipLaunchKernelGGL(pack_W_kernel, dim3((FF * NO + 255) / 256), dim3(256),
                           0, stream, W, Wp);
        // 64-row x 512-col blocks: (M/64) row blocks x 2 col halves
        hipLaunchKernelGGL(mg_gemm_packed, dim3((M / 64) * 2), dim3(256), 0, stream,
                           mg_emb, var_indices, adjc, Wp, bias, out);
    } else {
        hipLaunchKernelGGL(mg_gemm_direct, dim3(M / 16), dim3(256), 0, stream,
                           mg_emb, var_indices, adjc, W, bias, out);
    }
}